// StudentTrans_86509231276550
// MI455X (gfx1250) — compile-verified
//
#include <hip/hip_runtime.h>
#include <math.h>

// Problem constants (from reference)
static constexpr int kB  = 65536;  // batch
static constexpr int kQ  = 15;     // queries / experts
static constexpr int kD  = 128;    // decoder dim
static constexpr int kCH = 256;    // expert hidden
static constexpr int kE  = 15;     // experts

typedef float v2f __attribute__((ext_vector_type(2)));
typedef float v8f __attribute__((ext_vector_type(8)));

// LDS row strides (floats), padded so row*stride mod 64 banks varies (132%64==4, 260%64==4)
#define LDA 132
#define LDH 260

// ---------------------------------------------------------------------------
// Kernel 0: zero the per-expert bin counters in workspace
// ---------------------------------------------------------------------------
__global__ void zero_counts_kernel(int* __restrict__ counts) {
    if (threadIdx.x < 16) counts[threadIdx.x] = 0;
}

// ---------------------------------------------------------------------------
// Kernel 1: per-sample scene scores + argmax + expert binning.
// One wave32 per sample. 256 threads = 8 waves = 8 samples per block.
// ---------------------------------------------------------------------------
__global__ __launch_bounds__(256) void scene_argmax_bin_kernel(
    const float* __restrict__ descs,    // [B, Q, D]
    const float* __restrict__ sceneW,   // [1, D]
    const float* __restrict__ sceneB,   // [1]
    float* __restrict__ outScores,      // d_out + B*D, [B, Q]
    int* __restrict__ counts,           // [16]
    int* __restrict__ bins)             // [E, B] packed (b<<4)|q
{
    const int wave = threadIdx.x >> 5;
    const int lane = threadIdx.x & 31;
    const int b    = blockIdx.x * 8 + wave;
    if (b >= kB) return;

    // each lane holds 4 consecutive floats of scene_W (32 lanes * 4 = 128)
    const float4 sw = ((const float4*)sceneW)[lane];
    const float  bias = sceneB[0];
    const float4* row = (const float4*)(descs + (size_t)b * (kQ * kD));

    float myval = -INFINITY;   // lanes >= kQ stay -inf
    for (int q = 0; q < kQ; ++q) {
        float4 x = row[q * 32 + lane];         // 512B coalesced per wave
        float p = x.x * sw.x + x.y * sw.y + x.z * sw.z + x.w * sw.w;
        // wave32 all-reduce sum
        for (int m = 16; m; m >>= 1) p += __shfl_xor(p, m, 32);
        if (lane == q) myval = p;
    }
    if (lane < kQ) outScores[(size_t)b * kQ + lane] = myval + bias;

    // wave32 argmax, first-occurrence tie-break (matches jnp.argmax)
    float v = myval;
    int   idx = lane;
    for (int m = 16; m; m >>= 1) {
        float ov = __shfl_xor(v, m, 32);
        int   oi = __shfl_xor(idx, m, 32);
        if (ov > v || (ov == v && oi < idx)) { v = ov; idx = oi; }
    }
    if (lane == 0) {
        int pos = atomicAdd(&counts[idx], 1);
        bins[idx * kB + pos] = (b << 4) | idx;   // expert == winning query index
    }
}

// ---------------------------------------------------------------------------
// Kernel 2: grouped expert MLP via V_WMMA_F32_16X16X4_F32.
// grid = (tiles, E). Block = 256 threads = 8 waves. Each tile = 16 samples of
// one expert. Layer1: each wave owns 2 N-tiles (16 cols) of CH=256.
// Layer2: each wave owns 1 N-tile of D=128.
// ---------------------------------------------------------------------------
__global__ __launch_bounds__(256) void expert_mlp_kernel(
    const float* __restrict__ descs,  // [B, Q, D]
    const float* __restrict__ W1,     // [E, CH, D]
    const float* __restrict__ b1,     // [E, CH]
    const float* __restrict__ W2,     // [E, D, CH]
    const float* __restrict__ b2,     // [E, D]
    const int*   __restrict__ counts,
    const int*   __restrict__ bins,
    float* __restrict__ outVec)       // d_out, [B, D]
{
    __shared__ float sA[16 * LDA];    // gathered descriptors [16 x 128], padded
    __shared__ float sH[16 * LDH];    // hidden activations  [16 x 256], padded
    __shared__ int   sRow[16];        // sample index per tile row (-1 = invalid)

    const int e  = blockIdx.y;
    const int n  = counts[e];
    const int nt = (n + 15) >> 4;
    const float* W1e = W1 + (size_t)e * kCH * kD;
    const float* W2e = W2 + (size_t)e * kD * kCH;

    const int wave = threadIdx.x >> 5;
    const int lane = threadIdx.x & 31;
    const int lrow = lane & 15;        // matrix row (A) / col (B,C,D) within 16
    const int ak   = (lane >> 4) * 2;  // K sub-offset for A/B fragments

    for (int tile = blockIdx.x; tile < nt; tile += gridDim.x) {
        const int base = tile << 4;
        const int m    = min(16, n - base);

        // ---- stage gathered descriptors into LDS (float4, coalesced) ----
        for (int i = threadIdx.x; i < 512; i += 256) {       // 16 rows * 32 float4
            const int r  = i >> 5;
            const int c4 = i & 31;
            float4 val = make_float4(0.f, 0.f, 0.f, 0.f);
            if (r < m) {
                const int ent  = bins[e * kB + base + r];
                const int bsmp = ent >> 4;
                const int q    = ent & 15;
                val = ((const float4*)(descs + ((size_t)bsmp * kQ + q) * kD))[c4];
                if (c4 == 0) sRow[r] = bsmp;
            } else if (c4 == 0) {
                sRow[r] = -1;
            }
            *(float4*)&sA[r * LDA + c4 * 4] = val;
        }
        __syncthreads();

        // ---- layer 1: [16x128] x [128x256(T)] -> gelu -> sH ----
        {
            v8f c0 = {};
            v8f c1 = {};
            const int h0 = (wave * 2 + 0) * 16;
            const int h1 = (wave * 2 + 1) * 16;
            const float* w1a = W1e + (size_t)(h0 + lrow) * kD + ak; // B frag: W1[h,k]
            const float* w1b = W1e + (size_t)(h1 + lrow) * kD + ak;
            const float* ap  = &sA[lrow * LDA + ak];                // A frag from LDS
            #pragma unroll 8
            for (int k = 0; k < kD; k += 4) {
                v2f a  = *(const v2f*)(ap + k);
                v2f bA = *(const v2f*)(w1a + k);
                v2f bB = *(const v2f*)(w1b + k);
                c0 = __builtin_amdgcn_wmma_f32_16x16x4_f32(false, a, false, bA,
                                                           (short)0, c0, false, false);
                c1 = __builtin_amdgcn_wmma_f32_16x16x4_f32(false, a, false, bB,
                                                           (short)0, c1, false, false);
            }
            const float bias0 = b1[e * kCH + h0 + lrow];
            const float bias1 = b1[e * kCH + h1 + lrow];
            const int Mbase = (lane >> 4) * 8;
            #pragma unroll
            for (int v = 0; v < 8; ++v) {
                float x0 = c0[v] + bias0;
                float x1 = c1[v] + bias1;
                // exact-erf GELU (matches torch F.gelu default)
                x0 = 0.5f * x0 * (1.0f + erff(x0 * 0.70710678118654752440f));
                x1 = 0.5f * x1 * (1.0f + erff(x1 * 0.70710678118654752440f));
                sH[(Mbase + v) * LDH + h0 + lrow] = x0;
                sH[(Mbase + v) * LDH + h1 + lrow] = x1;
            }
        }
        __syncthreads();

        // ---- layer 2: [16x256] x [256x128(T)] -> scatter to d_out ----
        {
            v8f c = {};
            const int n0 = wave * 16;
            const float* w2p = W2e + (size_t)(n0 + lrow) * kCH + ak; // B frag: W2[d,k]
            const float* hp  = &sH[lrow * LDH + ak];                 // A frag from LDS
            #pragma unroll 8
            for (int k = 0; k < kCH; k += 4) {
                v2f a = *(const v2f*)(hp + k);
                v2f b = *(const v2f*)(w2p + k);
                c = __builtin_amdgcn_wmma_f32_16x16x4_f32(false, a, false, b,
                                                          (short)0, c, false, false);
            }
            const float bias = b2[e * kD + n0 + lrow];
            const int Mbase = (lane >> 4) * 8;
            #pragma unroll
            for (int v = 0; v < 8; ++v) {
                const int r = Mbase + v;
                const int bsmp = sRow[r];
                if (bsmp >= 0) outVec[(size_t)bsmp * kD + n0 + lrow] = c[v] + bias;
            }
        }
        __syncthreads();  // protect sA/sH/sRow before next tile iteration
    }
}

// ---------------------------------------------------------------------------
// Launch
// ---------------------------------------------------------------------------
extern "C" void kernel_launch(void* const* d_in, const int* in_sizes, int n_in,
                              void* d_out, int out_size, void* d_ws, size_t ws_size,
                              hipStream_t stream) {
    const float* descs  = (const float*)d_in[0];  // [B,Q,D]
    const float* sceneW = (const float*)d_in[1];  // [1,D]
    const float* sceneB = (const float*)d_in[2];  // [1]
    const float* W1     = (const float*)d_in[3];  // [E,CH,D]
    const float* b1     = (const float*)d_in[4];  // [E,CH]
    const float* W2     = (const float*)d_in[5];  // [E,D,CH]
    const float* b2     = (const float*)d_in[6];  // [E,D]

    float* outVec    = (float*)d_out;                       // [B,D]
    float* outScores = (float*)d_out + (size_t)kB * kD;     // [B,Q]

    // workspace: [0,64)  bin counters (16 ints)
    //            [64,..) bins: E * B ints (packed (b<<4)|q)
    int* counts = (int*)d_ws;
    int* bins   = (int*)((char*)d_ws + 64);

    zero_counts_kernel<<<1, 32, 0, stream>>>(counts);

    scene_argmax_bin_kernel<<<kB / 8, 256, 0, stream>>>(
        descs, sceneW, sceneB, outScores, counts, bins);

    dim3 grid2(256, kE);   // blocks loop over tiles, so any count works
    expert_mlp_kernel<<<grid2, 256, 0, stream>>>(
        descs, W1, b1, W2, b2, counts, bins, outVec);
}